// ECELoss_55327768707242
// MI455X (gfx1250) — compile-verified
//
#include <hip/hip_runtime.h>
#include <hip/hip_bf16.h>

#define N_BINS 10
#define THREADS 256
#define BLOCKS 1024
#define HIST_WORDS (THREADS * N_BINS)   // 2560 floats per histogram matrix

typedef __attribute__((ext_vector_type(2))) float v2f;
typedef __attribute__((ext_vector_type(4))) float v4f;
typedef __attribute__((ext_vector_type(8))) float v8f;

__device__ __forceinline__ void ece_elem(float x, float lab, float* conf, float* corr) {
    // sigmoid: p = 1 / (1 + exp(-x)) = 1 / (1 + exp2(-x*log2(e)))
    float e = __builtin_amdgcn_exp2f(x * -1.44269504088896340736f);
    float p = __builtin_amdgcn_rcpf(1.0f + e);
    // bin = ceil(p*10) - 1, clipped to [0,9]  (reference: lower < p <= upper)
    int bin = (int)__builtin_ceilf(p * 10.0f) - 1;
    bin = bin < 0 ? 0 : (bin > 9 ? 9 : bin);
    // accuracy: pred = (p > 0.5); correct = (pred == label), label in {0,1}
    float pred = (p > 0.5f) ? 1.0f : 0.0f;
    float c = (pred == lab) ? 1.0f : 0.0f;
    // per-lane private LDS bins: fire-and-forget ds_add_f32, conflict-free banks
    atomicAdd(&conf[bin], p);
    atomicAdd(&corr[bin], c);
}

__global__ __launch_bounds__(THREADS) void ece_hist_kernel(
    const float* __restrict__ logits, const float* __restrict__ labels,
    double* __restrict__ gacc, int n)
{
    __shared__ float h[2 * HIST_WORDS + 64];   // conf[256][10], corr[256][10], slack

    const int tid = threadIdx.x;
    for (int i = tid; i < 2 * HIST_WORDS + 64; i += THREADS) h[i] = 0.0f;
    __syncthreads();

    float* conf = h + tid * N_BINS;
    float* corr = h + HIST_WORDS + tid * N_BINS;

    const int gtid = (int)(blockIdx.x * THREADS + tid);
    const int nthreads = (int)(gridDim.x * THREADS);
    const int nvec = n >> 2;
    const v4f* lg4 = (const v4f*)logits;
    const v4f* lb4 = (const v4f*)labels;

    // streaming phase: non-temporal 128-bit loads, grid-stride
    for (int i = gtid; i < nvec; i += nthreads) {
        v4f x = __builtin_nontemporal_load(&lg4[i]);
        v4f l = __builtin_nontemporal_load(&lb4[i]);
        ece_elem(x[0], l[0], conf, corr);
        ece_elem(x[1], l[1], conf, corr);
        ece_elem(x[2], l[2], conf, corr);
        ece_elem(x[3], l[3], conf, corr);
    }
    // tail (n not divisible by 4)
    if (gtid == 0) {
        for (int i = nvec << 2; i < n; ++i) ece_elem(logits[i], labels[i], conf, corr);
    }

    __syncthreads();

    // Block reduction via WMMA: D = ones(16x4) * B(4x16) + C  ==> column sums.
    // Wave 0 reduces conf[256][10], wave 1 reduces corr[256][10].
    // With A = all-ones, D[m,n] = sum_k B[k,n] regardless of K-row ordering.
    const int wave = tid >> 5;                 // wave-uniform
    if (wave < 2) {
        const float* src = h + wave * HIST_WORDS;
        const int lane = tid & 31;
        const int ncol = lane & 15;            // D column / B column for this lane
        const int khi  = (lane >> 4) << 1;     // lanes 16-31 carry K rows +2

        v2f a; a[0] = 1.0f; a[1] = 1.0f;       // A = ones
        v8f acc = {};
        #pragma unroll 4
        for (int t = 0; t < THREADS; t += 4) {
            v2f b;
            b[0] = src[(t + khi) * N_BINS + ncol];       // garbage cols 10..15 stay
            b[1] = src[(t + khi + 1) * N_BINS + ncol];   // in their own D columns
            acc = __builtin_amdgcn_wmma_f32_16x16x4_f32(
                false, a, false, b, (short)0, acc, false, false);
        }
        // D VGPR0, lanes 0..15 = row M=0, column N=lane -> per-bin block totals
        if (lane < N_BINS) {
            atomicAdd(&gacc[wave * N_BINS + lane], (double)acc[0]);
        }
    }
}

__global__ void ece_zero_ws(double* g) {
    int t = threadIdx.x;
    if (t < 2 * N_BINS) g[t] = 0.0;
}

__global__ void ece_finalize(const double* __restrict__ g, float* __restrict__ out, int n) {
    if (threadIdx.x == 0) {
        // ECE = sum_b |sum_conf[b] - sum_correct[b]| / n   (empty bins contribute 0)
        double e = 0.0;
        for (int b = 0; b < N_BINS; ++b) {
            double d = g[b] - g[N_BINS + b];
            e += (d < 0.0) ? -d : d;
        }
        out[0] = (float)(e / (double)n);
    }
}

extern "C" void kernel_launch(void* const* d_in, const int* in_sizes, int n_in,
                              void* d_out, int out_size, void* d_ws, size_t ws_size,
                              hipStream_t stream) {
    const float* logits = (const float*)d_in[0];
    const float* labels = (const float*)d_in[1];
    const int n = in_sizes[0];
    double* gacc = (double*)d_ws;   // 20 doubles: [0..9]=sum_conf, [10..19]=sum_correct
    float* out = (float*)d_out;

    ece_zero_ws<<<1, 32, 0, stream>>>(gacc);
    ece_hist_kernel<<<BLOCKS, THREADS, 0, stream>>>(logits, labels, gacc, n);
    ece_finalize<<<1, 32, 0, stream>>>(gacc, out, n);
}